// SpectralCoherence_87024627351940
// MI455X (gfx1250) — compile-verified
//
#include <hip/hip_runtime.h>
#include <math.h>

typedef __attribute__((ext_vector_type(2))) float v2f;
typedef __attribute__((ext_vector_type(8))) float v8f;
typedef __attribute__((ext_vector_type(4))) int   v4i;

#define NK 256          // k1 points (n)
#define NG 201          // grid points (a, b)
#define NA_P 204        // a padded to multiple of 4  (WMMA K)
#define NB_P 208        // b padded to multiple of 16 (WMMA N / K)
#define ND 64           // Delta_y / Delta_z points (i, j)
#define NBT 13          // b tiles (208/16)
#define PI_F 3.14159265358979f

#define TAB_ELEMS (2 * ND * NA_P)            // cos+sin A tables: 26112 floats
#define LDS_BYTES ((TAB_ELEMS + 512) * 4)    // + g/w tables: 106496 bytes

#define AS1 __attribute__((address_space(1)))
#define AS3 __attribute__((address_space(3)))

#if __has_builtin(__builtin_amdgcn_global_load_async_to_lds_b128) && \
    __has_builtin(__builtin_amdgcn_s_wait_asynccnt)
#define HAVE_ASYNC_LDS 1
#else
#define HAVE_ASYNC_LDS 0
#endif

// workspace layout (float offsets) -- Phi is never materialized
static constexpr size_t OFF_G   = 0;                                  // 256
static constexpr size_t OFF_W   = 256;                                // 256
static constexpr size_t OFF_DEN = 512;                                // 256 (1/den)
static constexpr size_t OFF_F1P = 768;                                // NK*13 partials
static constexpr size_t OFF_CA  = OFF_F1P + 3584;                     // [ND][NA_P] cos*w2
static constexpr size_t OFF_SA  = OFF_CA + (size_t)ND * NA_P;         // [ND][NA_P] sin*w2
static constexpr size_t OFF_C3  = OFF_SA + (size_t)ND * NA_P;         // [NB_P][ND] cos*w3
static constexpr size_t OFF_S3  = OFF_C3 + (size_t)NB_P * ND;         // [NB_P][ND] -sin*w3
static constexpr size_t OFF_TR  = OFF_S3 + (size_t)NB_P * ND;         // [NK][ND][NB_P]
static constexpr size_t OFF_TI  = OFF_TR + (size_t)NK * ND * NB_P;    // [NK][ND][NB_P]

static __device__ __forceinline__ v8f wmma_f32(v2f a, v2f b, v8f c) {
    // D = A(16x4 f32) * B(4x16 f32) + C(16x16 f32), exact fp32 on the matrix pipe
    return __builtin_amdgcn_wmma_f32_16x16x4_f32(false, a, false, b, (short)0, c,
                                                 false, false);
}

// von Karman spectral tensor Phi11 -- branchless: always finite for our inputs
// (k1 >= 1e-3 so kk, s > 0; pad grid values are 0), select zeroes pad lanes.
static __device__ __forceinline__ float phi_eval(float k1, float k2, float k3,
                                                 float L, float Ts, float M,
                                                 bool valid) {
    float kk    = k1 * k1 + k2 * k2 + k3 * k3;
    float knorm = sqrtf(kk);
    float beta  = Ts * powf(L * knorm, -2.0f / 3.0f);
    float k30   = k3 + beta * k1;
    float kk0   = k1 * k1 + k2 * k2 + k30 * k30;
    float x     = L * sqrtf(kk0);
    float x2    = x * x;
    float E0    = M * (x2 * x2) / powf(1.0f + x2, 17.0f / 6.0f);
    float s     = k1 * k1 + k2 * k2;
    float C1    = beta * k1 * k1 * (kk0 - 2.0f * k30 * k30 + beta * k1 * k30)
                  / (kk * s);
    float C2    = k2 * kk0 / (s * sqrtf(s))
                  * atan2f(beta * k1 * sqrtf(s), kk0 - k30 * k1 * beta);
    float zeta1 = C1 - (k2 / k1) * C2;
    float phi   = E0 / (4.0f * PI_F * kk0 * kk0)
                  * (kk0 - k1 * k1 - 2.0f * k1 * k30 * zeta1 + s * zeta1 * zeta1);
    return valid ? phi : 0.0f;
}

// ---- grid + trapezoid weights -------------------------------------------
__global__ void sc_grid_kernel(float* ws) {
    __shared__ float sg[NG];
    int i = threadIdx.x;
    if (i < NG) {
        float v;
        if (i == 100) {
            v = 0.0f;
        } else {
            int k = (i < 100) ? (99 - i) : (i - 101);
            float e = -2.0f + 4.0f * (float)k / 99.0f;     // logspace(-2,2,100)
            float gp = powf(10.0f, 2.0f * e);              // (10^e)^2
            v = (i < 100) ? -gp : gp;
        }
        sg[i] = v;
    }
    __syncthreads();
    if (i < NG) {
        ws[OFF_G + i] = sg[i];
        float wv;
        if (i == 0)            wv = 0.5f * (sg[1] - sg[0]);
        else if (i == NG - 1)  wv = 0.5f * (sg[NG - 1] - sg[NG - 2]);
        else                   wv = 0.5f * (sg[i + 1] - sg[i - 1]);
        ws[OFF_W + i] = wv;
    } else {
        ws[OFF_G + i] = 0.0f;
        ws[OFF_W + i] = 0.0f;
    }
}

// ---- trig operand tables (trapz weights folded in, -sin folded for GEMM2) -
__global__ void sc_trig_kernel(float* ws, const float* __restrict__ dy,
                               const float* __restrict__ dz) {
    int t = blockIdx.x * blockDim.x + threadIdx.x;
    const int nA = ND * NA_P;      // cos/sin A tables, [i][a]
    const int nB = NB_P * ND;      // cos/-sin B tables, [b][j]
    if (t < nA) {
        int i = t / NA_P, a = t % NA_P;
        float c = 0.0f, s = 0.0f;
        if (a < NG) {
            float sv, cv;
            sincosf(ws[OFF_G + a] * dy[i], &sv, &cv);
            float w = ws[OFF_W + a];
            c = cv * w; s = sv * w;
        }
        ws[OFF_CA + t] = c;
        ws[OFF_SA + t] = s;
    } else if (t < nA + nB) {
        int u = t - nA;
        int b = u / ND, j = u % ND;
        float c = 0.0f, s = 0.0f;
        if (b < NG) {
            float sv, cv;
            sincosf(ws[OFF_G + b] * dz[j], &sv, &cv);
            float w = ws[OFF_W + b];
            c = cv * w; s = -sv * w;   // minus sign from Re(T1 * e^{i...})
        }
        ws[OFF_C3 + u] = c;
        ws[OFF_S3 + u] = s;
    }
}

// ---- fused Phi + GEMM1 + F1 partials -------------------------------------
// One wave owns all four 16-row M-tiles of one (n, b-tile). Each lane computes
// two Phi values per K-step in registers (never stored) and feeds them as the
// shared B-operand of 8 fp32 WMMAs. A-operand cos/sin tables (shared by ALL
// waves) are staged in LDS via async-to-LDS copies; per-iteration fetches
// become conflict-free ds_load_b64 (stride 204 words -> 64 distinct banks).
__global__ void sc_fused1_kernel(float* ws, const float* __restrict__ k1in,
                                 const float* __restrict__ logL,
                                 const float* __restrict__ logT,
                                 const float* __restrict__ logM) {
    extern __shared__ float smem[];
    float* sTab = smem;                 // [2][ND][NA_P]: cos*w2 then sin*w2
    float* sGW  = smem + TAB_ELEMS;     // [256] g, [256] w

    int tid = threadIdx.x;
#if HAVE_ASYNC_LDS
    {
        for (int e = tid * 4; e < TAB_ELEMS; e += 256 * 4)
            __builtin_amdgcn_global_load_async_to_lds_b128(
                (AS1 v4i*)(ws + OFF_CA + e), (AS3 v4i*)(sTab + e), 0, 0);
        if (tid < 128)
            __builtin_amdgcn_global_load_async_to_lds_b128(
                (AS1 v4i*)(ws + OFF_G + tid * 4), (AS3 v4i*)(sGW + tid * 4), 0, 0);
        __builtin_amdgcn_s_wait_asynccnt(0);
    }
#else
    for (int e = tid; e < TAB_ELEMS; e += 256) sTab[e] = ws[OFF_CA + e];
    for (int e = tid; e < 512; e += 256)       sGW[e]  = ws[OFF_G + e];
#endif
    __syncthreads();

    int wave = (blockIdx.x * blockDim.x + threadIdx.x) >> 5;
    int n  = wave / NBT;
    int nt = wave % NBT;
    int lane = threadIdx.x & 31;
    int half = lane >> 4, l = lane & 15;
    int Bcol = nt * 16 + l;                    // b index 0..207

    float L  = expf(logL[0]);
    float Ts = expf(logT[0]);
    float M  = expf(logM[0]);
    float k1 = k1in[n];
    float k3 = sGW[Bcol];                      // zero in pad region
    float wb = sGW[256 + Bcol];
    bool  bok = (Bcol < NG);

    const float* cA = sTab + (size_t)l * NA_P;                  // row l, m-tile 0
    const float* sA = sTab + (size_t)(ND * NA_P) + (size_t)l * NA_P;

    v8f accR[4] = {{}, {}, {}, {}};
    v8f accI[4] = {{}, {}, {}, {}};
    float f1 = 0.0f;

    for (int k0 = 0; k0 < NA_P; k0 += 4) {
        int ks = k0 + 2 * half;                // lanes 0-15: K,K+1; 16-31: K+2,K+3
        float g0 = sGW[ks];
        float g1 = sGW[ks + 1];
        float p0 = phi_eval(k1, g0, k3, L, Ts, M, bok && (ks < NG));
        float p1 = phi_eval(k1, g1, k3, L, Ts, M, bok && (ks + 1 < NG));
        f1 += (p0 * sGW[256 + ks] + p1 * sGW[256 + ks + 1]) * wb;

        v2f bb; bb.x = p0; bb.y = p1;
#pragma unroll
        for (int mt = 0; mt < 4; ++mt) {
            v2f ac = *(const v2f*)(cA + (size_t)(mt * 16) * NA_P + ks);
            v2f as = *(const v2f*)(sA + (size_t)(mt * 16) * NA_P + ks);
            accR[mt] = wmma_f32(ac, bb, accR[mt]);
            accI[mt] = wmma_f32(as, bb, accI[mt]);
        }
    }

    // deterministic wave32 reduction of the F1 partial
    for (int m = 16; m > 0; m >>= 1) f1 += __shfl_xor(f1, m, 32);
    if (lane == 0) ws[OFF_F1P + wave] = f1;

    float* TR = ws + OFF_TR + ((size_t)n * ND) * NB_P;
    float* TI = ws + OFF_TI + ((size_t)n * ND) * NB_P;
#pragma unroll
    for (int mt = 0; mt < 4; ++mt) {
#pragma unroll
        for (int r = 0; r < 8; ++r) {
            int Mrow = mt * 16 + r + 8 * half;
            TR[(size_t)Mrow * NB_P + Bcol] = accR[mt][r];
            TI[(size_t)Mrow * NB_P + Bcol] = accI[mt][r];
        }
    }
}

// ---- fixed-order reduction of 13 partials per n -> 1/den -----------------
__global__ void sc_f1red_kernel(float* ws) {
    int n = threadIdx.x;                       // one block of 256
    float s = 0.0f;
    for (int t = 0; t < NBT; ++t) s += ws[OFF_F1P + (size_t)n * NBT + t];
    ws[OFF_DEN + n] = 1.0f / fabsf(s);         // den = sqrt(F1^2)
}

// ---- GEMM2: out[n,i,j] = (TR*cosw + TI*(-sinw)) * (1/den[n]) -------------
__global__ void sc_gemm2_kernel(const float* __restrict__ ws, float* __restrict__ out) {
    int wave = (blockIdx.x * blockDim.x + threadIdx.x) >> 5;
    if (wave >= NK * 16) return;               // 4 x 4 tiles per n
    int n  = wave / 16;
    int t  = wave % 16;
    int mt = t / 4, nt = t % 4;
    int lane = threadIdx.x & 31;
    int half = lane >> 4, l = lane & 15;

    int Arow = mt * 16 + l;                    // i
    int Bcol = nt * 16 + l;                    // j
    const float* AR = ws + OFF_TR + ((size_t)n * ND + Arow) * NB_P;
    const float* AI = ws + OFF_TI + ((size_t)n * ND + Arow) * NB_P;
    const float* Bc = ws + OFF_C3 + Bcol;      // [b][j], stride ND
    const float* Bs = ws + OFF_S3 + Bcol;

    v8f acc = {};
    for (int k0 = 0; k0 < NB_P; k0 += 4) {
        int ks = k0 + 2 * half;
        v2f ar = *(const v2f*)(AR + ks);
        v2f ai = *(const v2f*)(AI + ks);
        v2f bc, bs;
        bc.x = Bc[(size_t)ks * ND]; bc.y = Bc[(size_t)(ks + 1) * ND];
        bs.x = Bs[(size_t)ks * ND]; bs.y = Bs[(size_t)(ks + 1) * ND];
        acc = wmma_f32(ar, bc, acc);
        acc = wmma_f32(ai, bs, acc);
    }
    float inv = ws[OFF_DEN + n];
    for (int r = 0; r < 8; ++r) {
        int Mrow = mt * 16 + r + 8 * half;
        out[((size_t)n * ND + Mrow) * ND + Bcol] = acc[r] * inv;
    }
}

extern "C" void kernel_launch(void* const* d_in, const int* in_sizes, int n_in,
                              void* d_out, int out_size, void* d_ws, size_t ws_size,
                              hipStream_t stream) {
    const float* k1   = (const float*)d_in[0];   // 256
    const float* dy   = (const float*)d_in[1];   // 64
    const float* dz   = (const float*)d_in[2];   // 64
    const float* logL = (const float*)d_in[3];   // 1
    const float* logT = (const float*)d_in[4];   // 1
    const float* logM = (const float*)d_in[5];   // 1
    float* ws  = (float*)d_ws;
    float* out = (float*)d_out;

    sc_grid_kernel<<<1, 256, 0, stream>>>(ws);

    int trigTot = ND * NA_P + NB_P * ND;
    sc_trig_kernel<<<(trigTot + 255) / 256, 256, 0, stream>>>(ws, dy, dz);

    // 3328 waves, 8 waves per 256-thread block, 106.5 KB dynamic LDS per block
    sc_fused1_kernel<<<(NK * NBT) / 8, 256, LDS_BYTES, stream>>>(ws, k1, logL,
                                                                 logT, logM);

    sc_f1red_kernel<<<1, 256, 0, stream>>>(ws);

    sc_gemm2_kernel<<<(NK * 16) / 8, 256, 0, stream>>>(ws, out);
}